// MF_73658689126810
// MI455X (gfx1250) — compile-verified
//
#include <hip/hip_runtime.h>

typedef __attribute__((ext_vector_type(2))) float v2f;
typedef __attribute__((ext_vector_type(8))) float v8f;

#define FDIM 128
#define HDIM 64
#define CDIM 40
#define NLAYERS 2
#define MAXDEG 5

static __device__ __forceinline__ int imin(int a, int b) { return a < b ? a : b; }

// LDS fragment-order swizzle for a KxH weight matrix:
// slot(k,n) = ((k>>2)*2 + ((k>>1)&1))*HDIM + n   (float2 slot), parity k&1 picks .x/.y
// Lane read at k-step kk: frag = ((kk>>2)*2 + (lane>>4))*HDIM + n  -> one ds_load_b64.
static __device__ __forceinline__ int wslot(int k, int n) {
    return (((k >> 2) * 2 + ((k >> 1) & 1)) * HDIM + n) * 2 + (k & 1);
}

// ---------------- degree kernels ----------------
__global__ void zero_i32_kernel(int* __restrict__ p, int n) {
    int i = blockIdx.x * blockDim.x + threadIdx.x;
    if (i < n) p[i] = 0;
}

__global__ void deg_count_kernel(const int* __restrict__ src, const int* __restrict__ dst,
                                 int* __restrict__ deg, int E) {
    int e = blockIdx.x * blockDim.x + threadIdx.x;
    if (e >= E) return;
    int s = src[e], d = dst[e];
    if (s != d) atomicAdd(&deg[d], 1);
}

// ---------------- input GEMM: x_first = relu(x @ W1 + b1) ----------------
// one wave -> 16 rows x 64 cols, WMMA f32 16x16x4, B fed from swizzled LDS
__global__ void gemm_in_kernel(const float* __restrict__ x, const float* __restrict__ W1,
                               const float* __restrict__ b1, float* __restrict__ out, int N) {
    __shared__ v2f Wl[(FDIM / 4) * 2 * HDIM];  // 32 KB, fragment order
    int tid = threadIdx.x;
    {
        float* Wf = (float*)Wl;
        for (int i = tid; i < FDIM * HDIM; i += blockDim.x) {
            int k = i >> 6, n = i & 63;
            Wf[wslot(k, n)] = W1[i];
        }
    }
    __syncthreads();

    int wave = tid >> 5;
    int lane = tid & 31;
    int m0 = (blockIdx.x * (blockDim.x >> 5) + wave) * 16;
    if (m0 >= N) return;  // wave-uniform

    int mcol  = lane & 15;               // N index within 16-tile; also A row index
    int lhalf = lane >> 4;               // 0 or 1
    int khalf = lhalf << 1;              // 0 or 2
    int rowhi = lhalf << 3;              // C/D row offset: 0 or 8
    int arow  = imin(m0 + mcol, N - 1);  // clamp for tail loads

    v8f acc[4] = {};
    const float* xrow = x + (size_t)arow * FDIM;
    for (int kk = 0; kk < FDIM; kk += 4) {
        v2f a = *(const v2f*)(xrow + kk + khalf);  // global_load_b64
        int fbase = ((kk >> 2) * 2 + lhalf) * HDIM + mcol;
#pragma unroll
        for (int nt = 0; nt < 4; ++nt) {
            v2f b = Wl[fbase + nt * 16];  // ds_load_b64, no shuffles
            acc[nt] = __builtin_amdgcn_wmma_f32_16x16x4_f32(
                false, a, false, b, (short)0, acc[nt], false, false);
        }
    }

    bool full = (m0 + 16 <= N);  // wave-uniform fast path
    float* orow = out + (size_t)(m0 + rowhi) * HDIM + mcol;
#pragma unroll
    for (int nt = 0; nt < 4; ++nt) {
        float bias = b1[nt * 16 + mcol];
        if (full) {
#pragma unroll
            for (int j = 0; j < 8; ++j) {
                float v = acc[nt][j] + bias;
                orow[(size_t)j * HDIM + nt * 16] = v > 0.f ? v : 0.f;
            }
        } else {
#pragma unroll
            for (int j = 0; j < 8; ++j) {
                int r = m0 + j + rowhi;
                if (r < N) {
                    float v = acc[nt][j] + bias;
                    out[(size_t)r * HDIM + nt * 16 + mcol] = v > 0.f ? v : 0.f;
                }
            }
        }
    }
}

// ---------------- agg = h (copy, float4) ----------------
__global__ void copy_f4_kernel(const float4* __restrict__ src, float4* __restrict__ dst, int n4) {
    int i = blockIdx.x * blockDim.x + threadIdx.x;
    if (i < n4) dst[i] = src[i];
}

// ---------------- edge scatter: agg[dst] += h[src] (atomics) ----------------
// 16 threads per edge, each handles 4 consecutive floats
__global__ void edge_scatter_kernel(const int* __restrict__ src, const int* __restrict__ dst,
                                    const float* __restrict__ h, float* __restrict__ agg,
                                    long long E) {
    long long idx = (long long)blockIdx.x * blockDim.x + threadIdx.x;
    long long e = idx >> 4;
    if (e >= E) return;
    int f = (int)(idx & 15) << 2;
    int s = src[e], d = dst[e];
    if (s == d) return;  // remove_self_loops
    const float4 v = *(const float4*)(h + (size_t)s * HDIM + f);
    float* p = agg + (size_t)d * HDIM + f;
    atomicAdd(p + 0, v.x);
    atomicAdd(p + 1, v.y);
    atomicAdd(p + 2, v.z);
    atomicAdd(p + 3, v.w);
}

// ---------------- degree-bucketed linear + fuse ----------------
// h_out[n] = agg[n] @ relW[deg[n]] + relb[deg[n]] + fuse * x_first[n]
// A fragments loaded ONCE into registers, reused across all 6 buckets.
__global__ void bucket_linear_kernel(const float* __restrict__ agg, const int* __restrict__ deg,
                                     const float* __restrict__ relWl,   // [6,64,64] (layer slice)
                                     const float* __restrict__ relbl,   // [6,64]
                                     const float* __restrict__ xfirst,
                                     const float* __restrict__ fuse, int layer,
                                     float* __restrict__ hout, int N) {
    __shared__ v2f Wl[(HDIM / 4) * 2 * HDIM];  // 16 KB, fragment order, reloaded per bucket
    int tid  = threadIdx.x;
    int wave = tid >> 5;
    int lane = tid & 31;
    int m0 = (blockIdx.x * (blockDim.x >> 5) + wave) * 16;
    bool active = (m0 < N);  // cannot early-return: barriers inside bucket loop

    int mcol  = lane & 15;
    int lhalf = lane >> 4;
    int khalf = lhalf << 1;
    int rowhi = lhalf << 3;
    int arow  = active ? imin(m0 + mcol, N - 1) : 0;

    // per-lane degree for its 8 C/D rows
    int degs[8];
#pragma unroll
    for (int j = 0; j < 8; ++j) {
        int r = m0 + j + rowhi;
        degs[j] = (active && r < N) ? imin(deg[r], MAXDEG) : -1;
    }

    // load all A fragments once (K=64 -> 16 k-steps -> 32 VGPRs)
    v2f afrag[HDIM / 4];
    const float* arowp = agg + (size_t)arow * HDIM;
#pragma unroll
    for (int kk4 = 0; kk4 < HDIM / 4; ++kk4)
        afrag[kk4] = *(const v2f*)(arowp + kk4 * 4 + khalf);

    v8f sel[4] = {};
    float fu = fuse[layer];

    for (int k = 0; k <= MAXDEG; ++k) {
        __syncthreads();
        {
            float* Wf = (float*)Wl;
            const float* Wk = relWl + k * HDIM * HDIM;
            for (int i = tid; i < HDIM * HDIM; i += blockDim.x) {
                int kr = i >> 6, n = i & 63;
                Wf[wslot(kr, n)] = Wk[i];
            }
        }
        __syncthreads();

        v8f acc[4] = {};
#pragma unroll
        for (int kk4 = 0; kk4 < HDIM / 4; ++kk4) {
            v2f a = afrag[kk4];
            int fbase = (kk4 * 2 + lhalf) * HDIM + mcol;
#pragma unroll
            for (int nt = 0; nt < 4; ++nt) {
                v2f b = Wl[fbase + nt * 16];
                acc[nt] = __builtin_amdgcn_wmma_f32_16x16x4_f32(
                    false, a, false, b, (short)0, acc[nt], false, false);
            }
        }
        // per-row bucket select (== all_out[deg, node_idx])
#pragma unroll
        for (int nt = 0; nt < 4; ++nt) {
            float bias = relbl[k * HDIM + nt * 16 + mcol];
#pragma unroll
            for (int j = 0; j < 8; ++j) {
                if (degs[j] == k) sel[nt][j] = acc[nt][j] + bias;
            }
        }
    }

    bool full = active && (m0 + 16 <= N);  // wave-uniform fast path
    if (full) {
        const float* xrow = xfirst + (size_t)(m0 + rowhi) * HDIM + mcol;
        float* orow = hout + (size_t)(m0 + rowhi) * HDIM + mcol;
#pragma unroll
        for (int nt = 0; nt < 4; ++nt) {
#pragma unroll
            for (int j = 0; j < 8; ++j) {
                size_t o = (size_t)j * HDIM + nt * 16;
                orow[o] = sel[nt][j] + fu * xrow[o];
            }
        }
    } else {
#pragma unroll
        for (int nt = 0; nt < 4; ++nt) {
            int n = nt * 16 + mcol;
#pragma unroll
            for (int j = 0; j < 8; ++j) {
                int r = m0 + j + rowhi;
                if (active && r < N)
                    hout[(size_t)r * HDIM + n] = sel[nt][j] + fu * xfirst[(size_t)r * HDIM + n];
            }
        }
    }
}

// ---------------- output GEMM + log_softmax ----------------
__global__ void out_softmax_kernel(const float* __restrict__ h, const float* __restrict__ Wout,
                                   const float* __restrict__ bout, float* __restrict__ out, int N) {
    __shared__ float Wl[HDIM * CDIM];  // 10 KB
    __shared__ float bl[CDIM];
    int tid = threadIdx.x;
    for (int i = tid; i < HDIM * CDIM; i += blockDim.x) Wl[i] = Wout[i];
    if (tid < CDIM) bl[tid] = bout[tid];
    __syncthreads();

    int n = blockIdx.x * blockDim.x + tid;
    if (n >= N) return;

    float hr[HDIM];
#pragma unroll
    for (int d = 0; d < HDIM; ++d) hr[d] = h[(size_t)n * HDIM + d];

    float lg[CDIM];
    float m = -3.4e38f;
#pragma unroll 4
    for (int c = 0; c < CDIM; ++c) {
        float s = bl[c];
        for (int d = 0; d < HDIM; ++d) s += hr[d] * Wl[d * CDIM + c];
        lg[c] = s;
        m = s > m ? s : m;
    }
    float sum = 0.f;
#pragma unroll
    for (int c = 0; c < CDIM; ++c) sum += __expf(lg[c] - m);
    float lse = __logf(sum) + m;
#pragma unroll
    for (int c = 0; c < CDIM; ++c) out[(size_t)n * CDIM + c] = lg[c] - lse;
}

extern "C" void kernel_launch(void* const* d_in, const int* in_sizes, int n_in,
                              void* d_out, int out_size, void* d_ws, size_t ws_size,
                              hipStream_t stream) {
    const float* x    = (const float*)d_in[0];
    const int*   ei   = (const int*)d_in[1];
    const float* W1   = (const float*)d_in[2];
    const float* b1   = (const float*)d_in[3];
    const float* relW = (const float*)d_in[4];  // [2,6,64,64]
    const float* relb = (const float*)d_in[5];  // [2,6,64]
    const float* Wout = (const float*)d_in[6];  // [64,40]
    const float* bout = (const float*)d_in[7];  // [40]
    const float* fuse = (const float*)d_in[8];  // [2]

    const int N = in_sizes[0] / FDIM;
    const long long E = in_sizes[1] / 2;
    const int* src = ei;
    const int* dst = ei + E;

    // workspace layout
    char* ws = (char*)d_ws;
    size_t off = 0;
    int* deg = (int*)(ws + off);
    off += (((size_t)N * sizeof(int)) + 255) & ~(size_t)255;
    float* xfirst = (float*)(ws + off);
    off += (size_t)N * HDIM * sizeof(float);
    float* hbuf = (float*)(ws + off);
    off += (size_t)N * HDIM * sizeof(float);
    float* agg = (float*)(ws + off);
    off += (size_t)N * HDIM * sizeof(float);
    (void)ws_size;

    // 1) degrees
    zero_i32_kernel<<<(N + 255) / 256, 256, 0, stream>>>(deg, N);
    deg_count_kernel<<<(unsigned)((E + 255) / 256), 256, 0, stream>>>(src, dst, deg, (int)E);

    // 2) x_first = relu(x @ W1 + b1)
    const int WAVES_PER_BLOCK = 8;  // 256 threads, 8 waves -> 128 rows per block
    dim3 blk(32 * WAVES_PER_BLOCK);
    unsigned grid_rows = (unsigned)((N + 16 * WAVES_PER_BLOCK - 1) / (16 * WAVES_PER_BLOCK));
    gemm_in_kernel<<<grid_rows, blk, 0, stream>>>(x, W1, b1, xfirst, N);

    // 3) layers
    const float* hcur = xfirst;
    const int n4 = N * HDIM / 4;
    for (int layer = 0; layer < NLAYERS; ++layer) {
        copy_f4_kernel<<<(n4 + 255) / 256, 256, 0, stream>>>((const float4*)hcur, (float4*)agg, n4);
        long long nthreads = E * 16;
        edge_scatter_kernel<<<(unsigned)((nthreads + 255) / 256), 256, 0, stream>>>(
            src, dst, hcur, agg, E);
        bucket_linear_kernel<<<grid_rows, blk, 0, stream>>>(
            agg, deg, relW + (size_t)layer * 6 * HDIM * HDIM, relb + (size_t)layer * 6 * HDIM,
            xfirst, fuse, layer, hbuf, N);
        hcur = hbuf;
    }

    // 4) logits + log_softmax
    out_softmax_kernel<<<(N + 127) / 128, 128, 0, stream>>>(hcur, Wout, bout, (float*)d_out, N);
}